// YoloV2Loss_85718957294197
// MI455X (gfx1250) — compile-verified
//
#include <hip/hip_runtime.h>
#include <hip/hip_bf16.h>

// YOLO-v2 loss, CDNA5 (gfx1250, wave32).
// Streaming bandwidth-bound pass: 1 thread = 1 cell (30 f32 from predict +
// 30 f32 from target, loaded as nontemporal b64s), then a wave32 reduction
// implemented with V_WMMA_F32_16X16X4_F32 (B = ones => row-sum engine),
// block partials to d_ws, deterministic single-block final reduce to d_out.

#define LCELL 30
#define CCLS  20

typedef float v2f __attribute__((ext_vector_type(2)));
typedef float v8f __attribute__((ext_vector_type(8)));

#if __has_builtin(__builtin_amdgcn_wmma_f32_16x16x4_f32)
#define HAVE_WMMA_F32X4 1
#else
#define HAVE_WMMA_F32X4 0
#endif

// Exact f32 sum of all 32 lanes' v, broadcast to every lane.
// WMMA path: A(16x4): lanes0-15 hold K=0 (vgpr0), lanes16-31 hold K=2;
// vgpr1 (K=1/3) = 0. B = all ones. D[m,n] = v(m)+v(16+m) for every n.
// s(lane)=sum of 8 accum regs => s(l)+s(l^16) = total over all 32 lanes.
__device__ __forceinline__ float wave_sum32(float v) {
#if HAVE_WMMA_F32X4
  v2f a; a[0] = v;    a[1] = 0.0f;
  v2f b; b[0] = 1.0f; b[1] = 1.0f;
  v8f c = {0.f, 0.f, 0.f, 0.f, 0.f, 0.f, 0.f, 0.f};
  c = __builtin_amdgcn_wmma_f32_16x16x4_f32(false, a, false, b, (short)0, c,
                                            false, false);
  float s = ((c[0] + c[1]) + (c[2] + c[3])) + ((c[4] + c[5]) + (c[6] + c[7]));
  s += __shfl_xor(s, 16, 32);
  return s;
#else
  for (int off = 16; off > 0; off >>= 1) v += __shfl_xor(v, off, 32);
  return v;
#endif
}

__global__ __launch_bounds__(256) void yolo_cells_kernel(
    const float* __restrict__ p, const float* __restrict__ t,
    float* __restrict__ partials, int ncells) {
  const int cell  = blockIdx.x * 256 + threadIdx.x;
  const int c     = cell < ncells ? cell : ncells - 1;   // keep EXEC full
  const float val = cell < ncells ? 1.0f : 0.0f;

  // 30 floats @ cell*120 bytes: 8-byte aligned -> b64 nontemporal loads.
  const v2f* __restrict__ p2 = (const v2f*)(p + (size_t)c * LCELL);
  const v2f* __restrict__ t2 = (const v2f*)(t + (size_t)c * LCELL);
  float pv[LCELL], tv[LCELL];
#pragma unroll
  for (int i = 0; i < LCELL / 2; ++i) {
    v2f a = __builtin_nontemporal_load(p2 + i);
    v2f b = __builtin_nontemporal_load(t2 + i);
    pv[2 * i] = a[0]; pv[2 * i + 1] = a[1];
    tv[2 * i] = b[0]; tv[2 * i + 1] = b[1];
  }

  const float t29   = tv[LCELL - 1];
  const float coord = (t29 > 0.0f) ? 1.0f : 0.0f;
  const float noobj = (t29 == 0.0f) ? 1.0f : 0.0f;

  // class loss (cols 0..19)
  float cls = 0.0f;
#pragma unroll
  for (int i = 0; i < CCLS; ++i) {
    float d = pv[i] - tv[i];
    cls = fmaf(d, d, cls);
  }
  // bug-faithful noobj columns 4 and 9 of the full 30-vector
  const float d4 = pv[4] - tv[4];
  const float d9 = pv[9] - tv[9];
  const float nol = d4 * d4 + d9 * d9;

  // boxes: cxcywh (w,h squared) -> xyxy
  float px1[2], py1[2], px2[2], py2[2], pa[2], pcf[2];
  float tx1[2], ty1[2], tx2[2], ty2[2], ta[2], tcf[2];
#pragma unroll
  for (int j = 0; j < 2; ++j) {
    const int o = CCLS + 5 * j;
    float cx = pv[o], cy = pv[o + 1], w = pv[o + 2], h = pv[o + 3];
    pcf[j] = pv[o + 4];
    float w2 = w * w, h2 = h * h;
    px1[j] = cx - 0.5f * w2; py1[j] = cy - 0.5f * h2;
    px2[j] = cx + 0.5f * w2; py2[j] = cy + 0.5f * h2;
    pa[j] = (px2[j] - px1[j]) * (py2[j] - py1[j]);   // as in reference
    cx = tv[o]; cy = tv[o + 1]; w = tv[o + 2]; h = tv[o + 3];
    tcf[j] = tv[o + 4];
    w2 = w * w; h2 = h * h;
    tx1[j] = cx - 0.5f * w2; ty1[j] = cy - 0.5f * h2;
    tx2[j] = cx + 0.5f * w2; ty2[j] = cy + 0.5f * h2;
    ta[j] = (tx2[j] - tx1[j]) * (ty2[j] - ty1[j]);
  }

  // pairwise IoU [pred j][targ k]
  float iou[2][2];
#pragma unroll
  for (int j = 0; j < 2; ++j) {
#pragma unroll
    for (int k = 0; k < 2; ++k) {
      float ltx = fmaxf(px1[j], tx1[k]);
      float lty = fmaxf(py1[j], ty1[k]);
      float rbx = fminf(px2[j], tx2[k]);
      float rby = fminf(py2[j], ty2[k]);
      float wd = fmaxf(rbx - ltx, 0.0f);
      float hd = fmaxf(rby - lty, 0.0f);
      float inter = wd * hd;
      iou[j][k] = inter / (pa[j] + ta[k] - inter);
    }
  }
  // argmax over pred axis (ties -> index 0), per target box
  const int m0 = (iou[1][0] > iou[0][0]) ? 1 : 0;
  const int m1 = (iou[1][1] > iou[0][1]) ? 1 : 0;
  const float wm0 = coord * (((m0 == 0) || (m1 == 0)) ? 1.0f : 0.0f);
  const float wm1 = coord * (((m0 == 1) || (m1 == 1)) ? 1.0f : 0.0f);
  const float wm[2] = {wm0, wm1};

  float xywh = 0.0f, cont = 0.0f;
#pragma unroll
  for (int j = 0; j < 2; ++j) {
    float dx1 = px1[j] - tx1[j], dy1 = py1[j] - ty1[j];
    float dx2 = px2[j] - tx2[j], dy2 = py2[j] - ty2[j];
    float dcf = pcf[j] - tcf[j];
    xywh += wm[j] * (dx1 * dx1 + dy1 * dy1 + dx2 * dx2 + dy2 * dy2);
    cont += wm[j] * (dcf * dcf);
  }

  float cellLoss =
      (5.0f * xywh + cont + 0.5f * (noobj * nol) + coord * cls) * val;

  // wave32 sum via WMMA, then 8-wave block sum to one partial per block
  float s = wave_sum32(cellLoss);
  __shared__ float wsum[8];
  const int wid = threadIdx.x >> 5;
  const int lane = threadIdx.x & 31;
  if (lane == 0) wsum[wid] = s;
  __syncthreads();
  if (threadIdx.x == 0) {
    float b = 0.0f;
#pragma unroll
    for (int i = 0; i < 8; ++i) b += wsum[i];
    partials[blockIdx.x] = b;
  }
}

__global__ __launch_bounds__(256) void yolo_reduce_kernel(
    const float* __restrict__ partials, int n, float* __restrict__ out) {
  float s = 0.0f;
  for (int i = threadIdx.x; i < n; i += 256) s += partials[i];
  s = wave_sum32(s);  // converged here: EXEC all ones
  __shared__ float wsum[8];
  const int wid = threadIdx.x >> 5;
  const int lane = threadIdx.x & 31;
  if (lane == 0) wsum[wid] = s;
  __syncthreads();
  if (threadIdx.x == 0) {
    float b = 0.0f;
#pragma unroll
    for (int i = 0; i < 8; ++i) b += wsum[i];
    out[0] = b;
  }
}

extern "C" void kernel_launch(void* const* d_in, const int* in_sizes, int n_in,
                              void* d_out, int out_size, void* d_ws,
                              size_t ws_size, hipStream_t stream) {
  const float* p = (const float*)d_in[0];   // predict [16384, 1470]
  const float* t = (const float*)d_in[1];   // target  [16384, 7,7,30]
  float* out = (float*)d_out;               // scalar f32 loss
  float* partials = (float*)d_ws;           // nblocks floats

  const int ncells = in_sizes[0] / LCELL;              // 802816
  const int nblocks = (ncells + 255) / 256;            // 3136 (exact)

  yolo_cells_kernel<<<nblocks, 256, 0, stream>>>(p, t, partials, ncells);
  yolo_reduce_kernel<<<1, 256, 0, stream>>>(partials, nblocks, out);
}